// CustomImageCosineSimLoss_63952063038062
// MI455X (gfx1250) — compile-verified
//
#include <hip/hip_runtime.h>
#include <hip/hip_bf16.h>

typedef float v2f __attribute__((ext_vector_type(2)));
typedef float v8f __attribute__((ext_vector_type(8)));

#define EPS_COS 1e-8f
#define EPS_W   1e-6f

// ---------------------------------------------------------------------------
// Kernel 1: per-row L2 norms of image & text features; also zero accumulator.
// One wave per row (wave32), 8 waves per block.
// ---------------------------------------------------------------------------
__global__ __launch_bounds__(256) void norms_kernel(
    const float* __restrict__ I, const float* __restrict__ T,
    float* __restrict__ imgn, float* __restrict__ txtn,
    float* __restrict__ acc, int N, int D) {
  const int lane = threadIdx.x & 31;
  const int wave = threadIdx.x >> 5;
  const int row  = blockIdx.x * 8 + wave;
  if (blockIdx.x == 0 && threadIdx.x == 0) *acc = 0.0f;
  if (row >= N) return;
  const float* pi = I + (size_t)row * D;
  const float* pt = T + (size_t)row * D;
  float si = 0.f, st = 0.f;
  for (int t = lane; t < D; t += 32) {
    float a = pi[t]; si += a * a;
    float b = pt[t]; st += b * b;
  }
  #pragma unroll
  for (int off = 16; off > 0; off >>= 1) {
    si += __shfl_xor(si, off, 32);
    st += __shfl_xor(st, off, 32);
  }
  if (lane == 0) {
    imgn[row] = sqrtf(si);
    txtn[row] = sqrtf(st);
  }
}

// ---------------------------------------------------------------------------
// Fragment loader for V_WMMA_F32_16X16X4_F32 operands.
// A (16x4, MxK) and B (4x16, KxN) share the same lane layout when both source
// matrices are row-major [row, d]:
//   lane L holds row = row0 + (L&15), K = 2*(L>>4) + {0,1}  -> one b64 load.
// ---------------------------------------------------------------------------
__device__ __forceinline__ const float* frag_base(const float* M, int row0,
                                                  int D, int lane) {
  return M + (size_t)(row0 + (lane & 15)) * D + ((lane >> 4) << 1);
}

#define WMMA_F32(A, B, C) \
  __builtin_amdgcn_wmma_f32_16x16x4_f32(false, (A), false, (B), (short)0, (C), false, false)

// ---------------------------------------------------------------------------
// Kernel 2: sim = T * T^T  (fp32 WMMA).  Wave tile: 32 rows x 64 cols.
// Block = 8 waves stacked in rows -> 256 x 64 per block.
// ---------------------------------------------------------------------------
__global__ __launch_bounds__(256) void sim_gemm_kernel(
    const float* __restrict__ T, float* __restrict__ sim, int N, int D) {
  const int lane = threadIdx.x & 31;
  const int wave = threadIdx.x >> 5;
  const int row0 = blockIdx.y * 256 + wave * 32;
  const int col0 = blockIdx.x * 64;

  const float* pa0 = frag_base(T, row0,      D, lane);
  const float* pa1 = frag_base(T, row0 + 16, D, lane);
  const float* pb0 = frag_base(T, col0,      D, lane);
  const float* pb1 = frag_base(T, col0 + 16, D, lane);
  const float* pb2 = frag_base(T, col0 + 32, D, lane);
  const float* pb3 = frag_base(T, col0 + 48, D, lane);

  v8f c[2][4] = {};
  for (int d = 0; d < D; d += 4) {
    v2f a0 = *(const v2f*)(pa0 + d);
    v2f a1 = *(const v2f*)(pa1 + d);
    v2f b0 = *(const v2f*)(pb0 + d);
    v2f b1 = *(const v2f*)(pb1 + d);
    v2f b2 = *(const v2f*)(pb2 + d);
    v2f b3 = *(const v2f*)(pb3 + d);
    c[0][0] = WMMA_F32(a0, b0, c[0][0]);
    c[0][1] = WMMA_F32(a0, b1, c[0][1]);
    c[0][2] = WMMA_F32(a0, b2, c[0][2]);
    c[0][3] = WMMA_F32(a0, b3, c[0][3]);
    c[1][0] = WMMA_F32(a1, b0, c[1][0]);
    c[1][1] = WMMA_F32(a1, b1, c[1][1]);
    c[1][2] = WMMA_F32(a1, b2, c[1][2]);
    c[1][3] = WMMA_F32(a1, b3, c[1][3]);
  }

  // C/D layout: VGPR r -> M = r (lanes 0-15) or r+8 (lanes 16-31), N = lane&15
  const int chi  = (lane >> 4) * 8;
  const int csel = lane & 15;
  #pragma unroll
  for (int ti = 0; ti < 2; ti++)
    #pragma unroll
    for (int tj = 0; tj < 4; tj++)
      #pragma unroll
      for (int r = 0; r < 8; r++) {
        int i = row0 + 16 * ti + r + chi;
        int j = col0 + 16 * tj + csel;
        sim[(size_t)i * N + j] = c[ti][tj][r];
      }
}

// ---------------------------------------------------------------------------
// Kernel 3: per-row min / max of sim. One block (256 threads) per row.
// ---------------------------------------------------------------------------
__global__ __launch_bounds__(256) void minmax_kernel(
    const float* __restrict__ sim, float* __restrict__ mn,
    float* __restrict__ mx, int N) {
  const int row  = blockIdx.x;
  const int lane = threadIdx.x & 31;
  const int wave = threadIdx.x >> 5;
  const float* p = sim + (size_t)row * N;
  float mnv = 3.402823466e38f, mxv = -3.402823466e38f;
  for (int j = threadIdx.x; j < N; j += 256) {
    float v = p[j];
    mnv = fminf(mnv, v);
    mxv = fmaxf(mxv, v);
  }
  #pragma unroll
  for (int off = 16; off > 0; off >>= 1) {
    mnv = fminf(mnv, __shfl_xor(mnv, off, 32));
    mxv = fmaxf(mxv, __shfl_xor(mxv, off, 32));
  }
  __shared__ float smn[8], smx[8];
  if (lane == 0) { smn[wave] = mnv; smx[wave] = mxv; }
  __syncthreads();
  if (threadIdx.x == 0) {
    float a = smn[0], b = smx[0];
    #pragma unroll
    for (int w = 1; w < 8; w++) { a = fminf(a, smn[w]); b = fmaxf(b, smx[w]); }
    mn[row] = a;
    mx[row] = b;
  }
}

// ---------------------------------------------------------------------------
// Kernel 4: fused loss. Recomputes dots = I * T^T tile in registers via WMMA,
// reads sim tile + norms + instr ids, evaluates the per-pair loss, and
// block-reduces into a single fp32 atomic per block.
// ---------------------------------------------------------------------------
__global__ __launch_bounds__(256) void loss_kernel(
    const float* __restrict__ I, const float* __restrict__ T,
    const float* __restrict__ sim, const float* __restrict__ imgn,
    const float* __restrict__ txtn, const float* __restrict__ mn,
    const float* __restrict__ mx, const int* __restrict__ instr,
    float* __restrict__ acc, int N, int D) {
  const int lane = threadIdx.x & 31;
  const int wave = threadIdx.x >> 5;
  const int row0 = blockIdx.y * 256 + wave * 32;
  const int col0 = blockIdx.x * 64;

  const float* pa0 = frag_base(I, row0,      D, lane);
  const float* pa1 = frag_base(I, row0 + 16, D, lane);
  const float* pb0 = frag_base(T, col0,      D, lane);
  const float* pb1 = frag_base(T, col0 + 16, D, lane);
  const float* pb2 = frag_base(T, col0 + 32, D, lane);
  const float* pb3 = frag_base(T, col0 + 48, D, lane);

  v8f c[2][4] = {};
  for (int d = 0; d < D; d += 4) {
    v2f a0 = *(const v2f*)(pa0 + d);
    v2f a1 = *(const v2f*)(pa1 + d);
    v2f b0 = *(const v2f*)(pb0 + d);
    v2f b1 = *(const v2f*)(pb1 + d);
    v2f b2 = *(const v2f*)(pb2 + d);
    v2f b3 = *(const v2f*)(pb3 + d);
    c[0][0] = WMMA_F32(a0, b0, c[0][0]);
    c[0][1] = WMMA_F32(a0, b1, c[0][1]);
    c[0][2] = WMMA_F32(a0, b2, c[0][2]);
    c[0][3] = WMMA_F32(a0, b3, c[0][3]);
    c[1][0] = WMMA_F32(a1, b0, c[1][0]);
    c[1][1] = WMMA_F32(a1, b1, c[1][1]);
    c[1][2] = WMMA_F32(a1, b2, c[1][2]);
    c[1][3] = WMMA_F32(a1, b3, c[1][3]);
  }

  const int chi  = (lane >> 4) * 8;
  const int csel = lane & 15;
  float lsum = 0.0f;
  #pragma unroll
  for (int ti = 0; ti < 2; ti++) {
    #pragma unroll
    for (int r = 0; r < 8; r++) {
      const int i = row0 + 16 * ti + r + chi;
      const float mni = mn[i];
      const float inv_den = 1.0f / (mx[i] - mni + EPS_W);
      const float ni = imgn[i];
      const int   qi = instr[i];
      #pragma unroll
      for (int tj = 0; tj < 4; tj++) {
        const int j   = col0 + 16 * tj + csel;
        const float dot  = c[ti][tj][r];
        const float s    = sim[(size_t)i * N + j];
        const float w    = (s - mni) * inv_den;
        const float cosv = dot / fmaxf(ni * txtn[j], EPS_COS);
        const bool  al   = (qi == instr[j]) || (i == j);
        lsum += al ? (1.0f - cosv) : fmaxf(0.0f, cosv - w);
      }
    }
  }

  #pragma unroll
  for (int off = 16; off > 0; off >>= 1) lsum += __shfl_xor(lsum, off, 32);
  __shared__ float sred[8];
  if (lane == 0) sred[wave] = lsum;
  __syncthreads();
  if (threadIdx.x == 0) {
    float s = 0.f;
    #pragma unroll
    for (int w = 0; w < 8; w++) s += sred[w];
    atomicAdd(acc, s);
  }
}

// ---------------------------------------------------------------------------
// Kernel 5: finalize mean.
// ---------------------------------------------------------------------------
__global__ void finalize_kernel(const float* __restrict__ acc,
                                float* __restrict__ out, int N) {
  out[0] = acc[0] / ((float)N * (float)N);
}

extern "C" void kernel_launch(void* const* d_in, const int* in_sizes, int n_in,
                              void* d_out, int out_size, void* d_ws, size_t ws_size,
                              hipStream_t stream) {
  const float* I    = (const float*)d_in[0];
  const float* T    = (const float*)d_in[1];
  const int*  instr = (const int*)d_in[2];

  const int N = in_sizes[2];            // 4096
  const int D = in_sizes[0] / N;        // 512

  char* ws = (char*)d_ws;
  float* sim  = (float*)ws;                         size_t off = (size_t)N * N * sizeof(float);
  float* imgn = (float*)(ws + off);                 off += (size_t)N * sizeof(float);
  float* txtn = (float*)(ws + off);                 off += (size_t)N * sizeof(float);
  float* mn   = (float*)(ws + off);                 off += (size_t)N * sizeof(float);
  float* mx   = (float*)(ws + off);                 off += (size_t)N * sizeof(float);
  float* acc  = (float*)(ws + off);

  dim3 gemm_grid(N / 64, N / 256);

  norms_kernel<<<N / 8, 256, 0, stream>>>(I, T, imgn, txtn, acc, N, D);
  sim_gemm_kernel<<<gemm_grid, 256, 0, stream>>>(T, sim, N, D);
  minmax_kernel<<<N, 256, 0, stream>>>(sim, mn, mx, N);
  loss_kernel<<<gemm_grid, 256, 0, stream>>>(I, T, sim, imgn, txtn, mn, mx,
                                             instr, acc, N, D);
  finalize_kernel<<<1, 1, 0, stream>>>(acc, (float*)d_out, N);
}